// PointNetPP_DAM_16930761080955
// MI455X (gfx1250) — compile-verified
//
#include <hip/hip_runtime.h>
#include <hip/hip_bf16.h>
#include <float.h>
#include <math.h>

typedef __attribute__((ext_vector_type(2))) float v2f;
typedef __attribute__((ext_vector_type(8))) float v8f;

#define B_  2
#define N_  16384
#define M_  4096
#define KG_ 32
#define RADIUS2 0.64f

static __device__ __forceinline__ v8f wmma_f32_k4(v2f a, v2f b, v8f c) {
  return __builtin_amdgcn_wmma_f32_16x16x4_f32(false, a, false, b, (short)0, c,
                                               false, false);
}

// ---------------------------------------------------------------------------
// Kernel 1: Farthest Point Sampling. One block per batch; whole cloud held in
// registers (16 pts/thread). Argmax per step: wave32 shfl_xor reduction +
// one cross-wave stage -> 3 barriers/step on the serial critical path.
// ---------------------------------------------------------------------------
#define FPS_T 1024
#define FPS_PPT (N_ / FPS_T)  // 16
__global__ __launch_bounds__(FPS_T) void fps_kernel(
    const float* __restrict__ xyz, int* __restrict__ idx_out,
    float* __restrict__ new_xyz) {
  const int b = blockIdx.x;
  const int t = threadIdx.x;
  const int lane = t & 31;
  const int wv = t >> 5;  // 32 waves
  const float* px = xyz + (size_t)b * N_ * 3;

  float lx[FPS_PPT], ly[FPS_PPT], lz[FPS_PPT], dmin[FPS_PPT];
#pragma unroll
  for (int i = 0; i < FPS_PPT; ++i) {
    int n = t + FPS_T * i;
    lx[i] = px[n * 3 + 0];
    ly[i] = px[n * 3 + 1];
    lz[i] = px[n * 3 + 2];
    dmin[i] = 1e10f;
  }

  __shared__ float sc[3];
  __shared__ float swv[32];
  __shared__ int swi[32];
  __shared__ int sfar;

  int far = 0;
  for (int m = 0; m < M_; ++m) {
    if (t == (far & (FPS_T - 1))) {
      const int slot = far >> 10;
      sc[0] = lx[slot]; sc[1] = ly[slot]; sc[2] = lz[slot];
    }
    __syncthreads();
    if (t == 0) {
      idx_out[b * M_ + m] = far;
      new_xyz[(size_t)(b * M_ + m) * 3 + 0] = sc[0];
      new_xyz[(size_t)(b * M_ + m) * 3 + 1] = sc[1];
      new_xyz[(size_t)(b * M_ + m) * 3 + 2] = sc[2];
    }
    const float cx = sc[0], cy = sc[1], cz = sc[2];
    float bv = -1.0f;
    int bi = 0;
#pragma unroll
    for (int i = 0; i < FPS_PPT; ++i) {
      float dx = lx[i] - cx, dy = ly[i] - cy, dz = lz[i] - cz;
      float d = dx * dx + dy * dy + dz * dz;
      dmin[i] = fminf(dmin[i], d);
      if (dmin[i] > bv) { bv = dmin[i]; bi = t + FPS_T * i; }
    }
    // wave32 reduction (max value, min index on ties = argmax-first semantics)
#pragma unroll
    for (int mask = 16; mask > 0; mask >>= 1) {
      float ov = __shfl_xor(bv, mask, 32);
      int oi = __shfl_xor(bi, mask, 32);
      if (ov > bv || (ov == bv && oi < bi)) { bv = ov; bi = oi; }
    }
    if (lane == 0) { swv[wv] = bv; swi[wv] = bi; }
    __syncthreads();
    if (wv == 0) {
      float v = swv[lane];
      int i = swi[lane];
#pragma unroll
      for (int mask = 16; mask > 0; mask >>= 1) {
        float ov = __shfl_xor(v, mask, 32);
        int oi = __shfl_xor(i, mask, 32);
        if (ov > v || (ov == v && oi < i)) { v = ov; i = oi; }
      }
      if (lane == 0) sfar = i;
    }
    __syncthreads();
    far = sfar;
  }
}

// ---------------------------------------------------------------------------
// Kernel 2: kNN top-32 (squared distances). 4 threads per centroid, each scans
// a contiguous quarter of N via LDS-staged chunks keeping a sorted top-32;
// stable 4-way merge (lexicographic (d,n) == top_k tie order) by thread q==0.
// ---------------------------------------------------------------------------
#define KNN_T 256
#define KNN_Q 4
#define KNN_CPB (KNN_T / KNN_Q)  // 64 centroids per block
#define KNN_CH 512
#define KNN_QN (N_ / KNN_Q)  // 4096 candidates per quarter
__global__ __launch_bounds__(KNN_T) void knn_kernel(
    const float* __restrict__ xyz, const float* __restrict__ new_xyz,
    int* __restrict__ nidx, float* __restrict__ nd2) {
  const int tid = threadIdx.x;
  const int q = tid & (KNN_Q - 1);
  const int cl = tid >> 2;
  const int gm = blockIdx.x * KNN_CPB + cl;  // 0..B*M-1
  const int b = gm / M_;
  const float* px = xyz + (size_t)b * N_ * 3;
  const float cx = new_xyz[(size_t)gm * 3 + 0];
  const float cy = new_xyz[(size_t)gm * 3 + 1];
  const float cz = new_xyz[(size_t)gm * 3 + 2];

  float kd[KG_];
  int kn[KG_];
#pragma unroll
  for (int j = 0; j < KG_; ++j) { kd[j] = FLT_MAX; kn[j] = 0; }
  float worst = FLT_MAX;

  __shared__ float sx[KNN_CH], sy[KNN_CH], sz[KNN_CH];
  __shared__ float lsd[KNN_CPB][KNN_Q - 1][KG_];
  __shared__ int lsn[KNN_CPB][KNN_Q - 1][KG_];

  for (int base = 0; base < N_; base += KNN_CH) {
    __syncthreads();
    for (int j = tid; j < KNN_CH; j += KNN_T) {
      sx[j] = px[(size_t)(base + j) * 3 + 0];
      sy[j] = px[(size_t)(base + j) * 3 + 1];
      sz[j] = px[(size_t)(base + j) * 3 + 2];
    }
    __syncthreads();
    if ((base / KNN_QN) == q) {  // this chunk belongs to my quarter
      for (int j = 0; j < KNN_CH; ++j) {
        float dx = sx[j] - cx, dy = sy[j] - cy, dz = sz[j] - cz;
        float d = dx * dx + dy * dy + dz * dz;
        if (d < worst) {
          int p = KG_ - 1;
          while (p > 0 && kd[p - 1] > d) {
            kd[p] = kd[p - 1]; kn[p] = kn[p - 1]; --p;
          }
          kd[p] = d;
          kn[p] = base + j;
          worst = kd[KG_ - 1];
        }
      }
    }
  }
  if (q > 0) {
#pragma unroll
    for (int j = 0; j < KG_; ++j) {
      lsd[cl][q - 1][j] = kd[j];
      lsn[cl][q - 1][j] = kn[j];
    }
  }
  __syncthreads();
  if (q == 0) {
    int p[KNN_Q] = {0, 0, 0, 0};
    for (int o = 0; o < KG_; ++o) {
      float bd = (p[0] < KG_) ? kd[p[0]] : FLT_MAX;
      int bn = (p[0] < KG_) ? kn[p[0]] : 0x7fffffff;
      int bq = 0;
#pragma unroll
      for (int qq = 1; qq < KNN_Q; ++qq) {
        float hd = (p[qq] < KG_) ? lsd[cl][qq - 1][p[qq]] : FLT_MAX;
        int hn = (p[qq] < KG_) ? lsn[cl][qq - 1][p[qq]] : 0x7fffffff;
        if (hd < bd || (hd == bd && hn < bn)) { bd = hd; bn = hn; bq = qq; }
      }
      p[bq]++;
      nd2[(size_t)gm * KG_ + o] = bd;
      nidx[(size_t)gm * KG_ + o] = bn;
    }
  }
}

// ---------------------------------------------------------------------------
// Kernel 3: grouped SharedMLP (3->32->32->64) + masked max, one WAVE per group.
// All layers on V_WMMA_F32_16X16X4_F32. Fragment builds are branch-free:
// activations zero-padded to K=4 and weights staged (padded) in LDS, so each
// A/B element is an unconditional LDS load at offset 4k+2*hi(+1) -> b64 pairs.
// ---------------------------------------------------------------------------
#define MLP_WAVES 2
__global__ __launch_bounds__(32 * MLP_WAVES) void groupmlp_kernel(
    const float* __restrict__ xyz, const float* __restrict__ new_xyz,
    const int* __restrict__ nidx, const float* __restrict__ nd2,
    const float* __restrict__ w1, const float* __restrict__ s1,
    const float* __restrict__ b1, const float* __restrict__ w2,
    const float* __restrict__ s2, const float* __restrict__ b2,
    const float* __restrict__ w3, const float* __restrict__ s3,
    const float* __restrict__ b3, float* __restrict__ FLout) {
  const int tid = threadIdx.x;
  const int lane = tid & 31;
  const int wv = tid >> 5;
  const int gm = blockIdx.x * MLP_WAVES + wv;  // 0..B*M-1
  const int b = gm / M_;

  __shared__ float bufA[MLP_WAVES][KG_][32];  // g (cols 0..3, col3=0) / L2 out
  __shared__ float bufB[MLP_WAVES][KG_][64];  // L1 out / L3 out
  __shared__ float smk[MLP_WAVES][KG_];       // squared neighbor distances
  __shared__ float sw1[32][4];                // w1 padded to K=4
  __shared__ float sw2[32][32];
  __shared__ float sw3[64][32];
  __shared__ float sp1[64], sp2[64], sp3[128];  // {scale||bias} per layer

  // ---- block-wide weight staging ----
  for (int i = tid; i < 32; i += 32 * MLP_WAVES) {
    sw1[i][0] = w1[i * 3 + 0];
    sw1[i][1] = w1[i * 3 + 1];
    sw1[i][2] = w1[i * 3 + 2];
    sw1[i][3] = 0.0f;
    sp1[i] = s1[i];
    sp1[32 + i] = b1[i];
    sp2[i] = s2[i];
    sp2[32 + i] = b2[i];
  }
  for (int i = tid; i < 64; i += 32 * MLP_WAVES) {
    sp3[i] = s3[i];
    sp3[64 + i] = b3[i];
  }
  for (int i = tid; i < 32 * 32; i += 32 * MLP_WAVES)
    sw2[i >> 5][i & 31] = w2[i];
  for (int i = tid; i < 64 * 32; i += 32 * MLP_WAVES)
    sw3[i >> 5][i & 31] = w3[i];

  float (*A)[32] = bufA[wv];
  float (*Bu)[64] = bufB[wv];
  float* mk = smk[wv];

  {  // load + center neighborhood (one row per lane), pad col3 with 0
    const int j = lane;
    const int n = nidx[(size_t)gm * KG_ + j];
    const float* p = xyz + ((size_t)b * N_ + n) * 3;
    A[j][0] = p[0] - new_xyz[(size_t)gm * 3 + 0];
    A[j][1] = p[1] - new_xyz[(size_t)gm * 3 + 1];
    A[j][2] = p[2] - new_xyz[(size_t)gm * 3 + 2];
    A[j][3] = 0.0f;
    mk[j] = nd2[(size_t)gm * KG_ + j];
  }
  __syncthreads();

  const int lo = lane & 15;
  const int hi2 = (lane >> 4) << 1;  // 0 or 2: first K index held by this lane

  // ---- layer 1: (32 x 3) @ w1^T -> 32 ch (K padded to 4) ----
#pragma unroll
  for (int rt = 0; rt < 2; ++rt) {
    const int row = lo + 16 * rt;
    v2f a;
    a.x = A[row][hi2];
    a.y = A[row][hi2 + 1];
#pragma unroll
    for (int nt = 0; nt < 2; ++nt) {
      const int oc = lo + 16 * nt;
      v2f wb;
      wb.x = sw1[oc][hi2];
      wb.y = sw1[oc][hi2 + 1];
      v8f c = {};
      c = wmma_f32_k4(a, wb, c);
      const float sc = sp1[oc], bi = sp1[32 + oc];
#pragma unroll
      for (int r = 0; r < 8; ++r) {
        const int orow = r + 4 * hi2 + 16 * rt;
        float v = c[r] * sc + bi;
        Bu[orow][oc] = v > 0.0f ? v : 0.0f;
      }
    }
  }
  __syncthreads();

  // ---- layer 2: 32 -> 32 (K=32 as 8 chained K=4 WMMAs) ----
#pragma unroll
  for (int rt = 0; rt < 2; ++rt) {
    const int row = lo + 16 * rt;
#pragma unroll
    for (int nt = 0; nt < 2; ++nt) {
      const int oc = lo + 16 * nt;
      v8f c = {};
#pragma unroll
      for (int k = 0; k < 8; ++k) {
        const int kb = 4 * k + hi2;
        v2f a, wb;
        a.x = Bu[row][kb];
        a.y = Bu[row][kb + 1];
        wb.x = sw2[oc][kb];
        wb.y = sw2[oc][kb + 1];
        c = wmma_f32_k4(a, wb, c);
      }
      const float sc = sp2[oc], bi = sp2[32 + oc];
#pragma unroll
      for (int r = 0; r < 8; ++r) {
        const int orow = r + 4 * hi2 + 16 * rt;
        float v = c[r] * sc + bi;
        A[orow][oc] = v > 0.0f ? v : 0.0f;
      }
    }
  }
  __syncthreads();

  // ---- layer 3: 32 -> 64 + radius mask ----
#pragma unroll
  for (int rt = 0; rt < 2; ++rt) {
    const int row = lo + 16 * rt;
#pragma unroll
    for (int nt = 0; nt < 4; ++nt) {
      const int oc = lo + 16 * nt;
      v8f c = {};
#pragma unroll
      for (int k = 0; k < 8; ++k) {
        const int kb = 4 * k + hi2;
        v2f a, wb;
        a.x = A[row][kb];
        a.y = A[row][kb + 1];
        wb.x = sw3[oc][kb];
        wb.y = sw3[oc][kb + 1];
        c = wmma_f32_k4(a, wb, c);
      }
      const float sc = sp3[oc], bi = sp3[64 + oc];
#pragma unroll
      for (int r = 0; r < 8; ++r) {
        const int orow = r + 4 * hi2 + 16 * rt;
        float v = c[r] * sc + bi;
        v = v > 0.0f ? v : 0.0f;
        Bu[orow][oc] = (mk[orow] <= RADIUS2) ? v : -INFINITY;
      }
    }
  }
  __syncthreads();

  // ---- masked max over KG neighbors ----
  for (int ch = lane; ch < 64; ch += 32) {
    float mx = -INFINITY;
#pragma unroll
    for (int j = 0; j < KG_; ++j) mx = fmaxf(mx, Bu[j][ch]);
    FLout[(size_t)gm * 64 + ch] = mx;
  }
}

// ---------------------------------------------------------------------------
// Kernel 4: adaptive threshold net on constant density (per-batch scalar).
// ---------------------------------------------------------------------------
__global__ __launch_bounds__(64) void thr_kernel(
    const float* __restrict__ tw1, const float* __restrict__ tb1,
    const float* __restrict__ tw2, const float* __restrict__ tb2,
    const float* __restrict__ tw3, const float* __restrict__ tb3,
    float* __restrict__ thr) {
  __shared__ float t1[64], t2[64];
  const int t = threadIdx.x;
  const float dens = 64.0f / (4.0f / 3.0f * 3.14159f * 1.0f);
  float v = tw1[t] * dens + tb1[t];
  t1[t] = v > 0.0f ? v : 0.0f;
  __syncthreads();
  float acc = tb2[t];
  for (int j = 0; j < 64; ++j) acc += tw2[t * 64 + j] * t1[j];
  t2[t] = acc > 0.0f ? acc : 0.0f;
  __syncthreads();
  if (t == 0) {
    float a = tb3[0];
    for (int j = 0; j < 64; ++j) a += tw3[j] * t2[j];
    const float sig = 1.0f / (1.0f + expf(-a));
    const float th = 20.0f + 40.0f * sig;
    thr[0] = th;
    thr[1] = th;
  }
}

// ---------------------------------------------------------------------------
// Kernel 5: DAM fusion, one wave per sampled point; writes (B,128,M) output.
// ---------------------------------------------------------------------------
#define FUS_WAVES 8
__global__ __launch_bounds__(32 * FUS_WAVES) void fuse_kernel(
    const float* __restrict__ pts_cam, const float* __restrict__ FI,
    const float* __restrict__ new_xyz, const int* __restrict__ idx,
    const float* __restrict__ FLbuf, const float* __restrict__ thr,
    const float* __restrict__ gw_raw, const float* __restrict__ gb_raw,
    const float* __restrict__ gw_img, const float* __restrict__ gb_img,
    const float* __restrict__ gw_lid, const float* __restrict__ gb_lid,
    const float* __restrict__ uw, const float* __restrict__ ub,
    const float* __restrict__ vw, const float* __restrict__ vb,
    float* __restrict__ out) {
  const int lane = threadIdx.x & 31;
  const int wv = threadIdx.x >> 5;
  const int gm = blockIdx.x * FUS_WAVES + wv;
  const int b = gm / M_;
  const int m = gm % M_;

  __shared__ float sFI[FUS_WAVES][64];
  __shared__ float sFL[FUS_WAVES][64];
  __shared__ float sS[FUS_WAVES][64];

  const int i0 = idx[gm];
  for (int c = lane; c < 64; c += 32) {
    sFI[wv][c] = FI[((size_t)b * 64 + c) * N_ + i0];
    sFL[wv][c] = FLbuf[(size_t)gm * 64 + c];
  }
  __syncthreads();

  const float nx = new_xyz[(size_t)gm * 3 + 0];
  const float ny = new_xyz[(size_t)gm * 3 + 1];
  const float nz = new_xyz[(size_t)gm * 3 + 2];

  for (int c = lane; c < 64; c += 32) {
    float fr = gb_raw[c] + gw_raw[c * 3 + 0] * nx + gw_raw[c * 3 + 1] * ny +
               gw_raw[c * 3 + 2] * nz;
    fr = fr > 0.0f ? fr : 0.0f;
    float fi = gb_img[c];
    float fl = gb_lid[c];
    for (int j = 0; j < 64; ++j) {
      fi += gw_img[c * 64 + j] * sFI[wv][j];
      fl += gw_lid[c * 64 + j] * sFL[wv][j];
    }
    fi = fi > 0.0f ? fi : 0.0f;
    fl = fl > 0.0f ? fl : 0.0f;
    sS[wv][c] = tanhf(fr + fi + fl);
  }
  __syncthreads();

  float du = ub[0], dv = vb[0];
  for (int j = 0; j < 64; ++j) {
    du += uw[j] * sS[wv][j];
    dv += vw[j] * sS[wv][j];
  }
  const float wI = 1.0f / (1.0f + expf(-du));
  const float wL = 1.0f / (1.0f + expf(-dv));

  const float z = pts_cam[((size_t)b * N_ + i0) * 3 + 2];
  const bool near = z <= thr[b];

  for (int c = lane; c < 64; c += 32) {
    const float FLc = sFL[wv][c];
    const float FIc = sFI[wv][c];
    out[((size_t)b * 128 + c) * M_ + m] = near ? FLc : FIc;
    out[((size_t)b * 128 + 64 + c) * M_ + m] = near ? (FIc * wI) : (FLc * wL);
  }
}

// ---------------------------------------------------------------------------
extern "C" void kernel_launch(void* const* d_in, const int* in_sizes, int n_in,
                              void* d_out, int out_size, void* d_ws,
                              size_t ws_size, hipStream_t stream) {
  const float* xyz = (const float*)d_in[0];
  const float* pts_cam = (const float*)d_in[1];
  const float* FI = (const float*)d_in[2];
  const float* w1 = (const float*)d_in[3];
  const float* s1 = (const float*)d_in[4];
  const float* b1 = (const float*)d_in[5];
  const float* w2 = (const float*)d_in[6];
  const float* s2 = (const float*)d_in[7];
  const float* b2 = (const float*)d_in[8];
  const float* w3 = (const float*)d_in[9];
  const float* s3 = (const float*)d_in[10];
  const float* b3 = (const float*)d_in[11];
  const float* tw1 = (const float*)d_in[12];
  const float* tb1 = (const float*)d_in[13];
  const float* tw2 = (const float*)d_in[14];
  const float* tb2 = (const float*)d_in[15];
  const float* tw3 = (const float*)d_in[16];
  const float* tb3 = (const float*)d_in[17];
  const float* gw_raw = (const float*)d_in[18];
  const float* gb_raw = (const float*)d_in[19];
  const float* gw_img = (const float*)d_in[20];
  const float* gb_img = (const float*)d_in[21];
  const float* gw_lid = (const float*)d_in[22];
  const float* gb_lid = (const float*)d_in[23];
  const float* uw = (const float*)d_in[24];
  const float* ub = (const float*)d_in[25];
  const float* vw = (const float*)d_in[26];
  const float* vb = (const float*)d_in[27];
  float* out = (float*)d_out;

  char* ws = (char*)d_ws;
  int* idx = (int*)(ws + 0);                       // B*M ints
  float* nxyz = (float*)(ws + 32768);              // B*M*3 f32
  int* nidx = (int*)(ws + 131072);                 // B*M*KG ints
  float* nd2 = (float*)(ws + 131072 + 1048576);    // B*M*KG f32
  float* FLbuf = (float*)(ws + 131072 + 2097152);  // B*M*64 f32
  float* thr = (float*)(ws + 131072 + 2097152 + 2097152);  // B f32

  fps_kernel<<<B_, FPS_T, 0, stream>>>(xyz, idx, nxyz);
  knn_kernel<<<(B_ * M_) / KNN_CPB, KNN_T, 0, stream>>>(xyz, nxyz, nidx, nd2);
  groupmlp_kernel<<<(B_ * M_) / MLP_WAVES, 32 * MLP_WAVES, 0, stream>>>(
      xyz, nxyz, nidx, nd2, w1, s1, b1, w2, s2, b2, w3, s3, b3, FLbuf);
  thr_kernel<<<1, 64, 0, stream>>>(tw1, tb1, tw2, tb2, tw3, tb3, thr);
  fuse_kernel<<<(B_ * M_) / FUS_WAVES, 32 * FUS_WAVES, 0, stream>>>(
      pts_cam, FI, nxyz, idx, FLbuf, thr, gw_raw, gb_raw, gw_img, gb_img,
      gw_lid, gb_lid, uw, ub, vw, vb, out);
}